// Compl_77558519431300
// MI455X (gfx1250) — compile-verified
//
#include <hip/hip_runtime.h>
#include <hip/hip_bf16.h>
#include <math.h>

#define H 64
#define NI 1535
#define NN 1536
#define BB 32
#define BN_EPS 1e-5f

typedef float v2f __attribute__((ext_vector_type(2)));
typedef float v8f __attribute__((ext_vector_type(8)));

__device__ __forceinline__ v8f wmma_f32_k4(v2f a, v2f b, v8f c) {
  // D = A(16x4 f32) * B(4x16 f32) + C(16x16 f32)
  return __builtin_amdgcn_wmma_f32_16x16x4_f32(false, a, false, b, (short)0, c, false, false);
}

// ---------------------------------------------------------------------------
// Kernel 1: encode all B*N input nodes:
//   h = relu(relu([x,y] @ W1^T + b1) @ W2^T + b2)   -> nodes[b][n][h]
// One wave per 16-row M-tile, 8 waves per block. fp32 WMMA 16x16x4.
// ---------------------------------------------------------------------------
__global__ void __launch_bounds__(256) encode_kernel(
    const float* __restrict__ xx, const float* __restrict__ yy,
    const float* __restrict__ W1, const float* __restrict__ b1,
    const float* __restrict__ W2, const float* __restrict__ b2,
    float* __restrict__ nodes)
{
  __shared__ float h1[8][16][H];   // per-wave fc1 output staging (32 KB)
  const int wave  = threadIdx.x >> 5;
  const int lane  = threadIdx.x & 31;
  const int lhalf = lane >> 4;     // 0: K+0/1 half, 1: K+2/3 half
  const int l15   = lane & 15;
  const int tileBase  = (blockIdx.x * 8 + wave) * 16;
  const int totalRows = BB * NI;

  // A-fragment source row for this lane (clamped; garbage rows are masked on store)
  int rowA = tileBase + l15;
  if (rowA >= totalRows) rowA = totalRows - 1;
  const int  bA = rowA / NI, nA = rowA % NI;
  const long xbase = (long)(bA * NI + nA) * 8;
  const float yval = yy[bA * NI + nA];

  // ---- GEMM1: [16 x 9(pad12)] @ [12 x 64] ----
  for (int nt = 0; nt < 4; ++nt) {
    v8f c = {};
    #pragma unroll
    for (int s = 0; s < 3; ++s) {
      const int kb = 4 * s + 2 * lhalf;
      const int k1 = kb + 1;
      v2f a, bf;
      a.x = (kb < 8) ? xx[xbase + kb] : ((kb == 8) ? yval : 0.f);
      a.y = (k1 < 8) ? xx[xbase + k1] : ((k1 == 8) ? yval : 0.f);
      const int nG = nt * 16 + l15;
      bf.x = (kb < 9) ? W1[nG * 9 + kb] : 0.f;   // B[k][n] = W1[n][k]
      bf.y = (k1 < 9) ? W1[nG * 9 + k1] : 0.f;
      c = wmma_f32_k4(a, bf, c);
    }
    #pragma unroll
    for (int r = 0; r < 8; ++r) {            // documented C layout
      const int m  = r + 8 * lhalf;
      const int nG = nt * 16 + l15;
      const float v = c[r] + b1[nG];
      h1[wave][m][nG] = v > 0.f ? v : 0.f;
    }
  }
  // per-wave LDS region; in-wave RAW dependency handled by compiler waits

  // ---- GEMM2: [16 x 64] @ [64 x 64] ----
  for (int nt = 0; nt < 4; ++nt) {
    v8f c = {};
    #pragma unroll
    for (int s = 0; s < 16; ++s) {
      const int kb = 4 * s + 2 * lhalf;
      v2f a, bf;
      a.x = h1[wave][l15][kb];
      a.y = h1[wave][l15][kb + 1];
      const int nG = nt * 16 + l15;
      bf.x = W2[nG * H + kb];                 // B[k][n] = W2[n][k]
      bf.y = W2[nG * H + kb + 1];
      c = wmma_f32_k4(a, bf, c);
    }
    #pragma unroll
    for (int r = 0; r < 8; ++r) {
      const int m   = r + 8 * lhalf;
      const int row = tileBase + m;
      if (row < totalRows) {
        const int bI = row / NI, nI = row % NI;
        const int nG = nt * 16 + l15;
        const float v = c[r] + b2[nG];
        nodes[((long)bI * NN + nI) * H + nG] = v > 0.f ? v : 0.f;
      }
    }
  }
}

// ---------------------------------------------------------------------------
// Kernel 2: output-node encoder: nodes[b][N][h] = relu(output_xx[b] @ Wenc^T + benc)
// ---------------------------------------------------------------------------
__global__ void outenc_kernel(const float* __restrict__ oxx,
                              const float* __restrict__ Wenc,
                              const float* __restrict__ benc,
                              float* __restrict__ nodes)
{
  const int t = blockIdx.x * blockDim.x + threadIdx.x;
  if (t >= BB * H) return;
  const int b = t / H, h = t % H;
  float acc = benc[h];
  #pragma unroll
  for (int k = 0; k < 8; ++k) acc += oxx[b * 8 + k] * Wenc[h * 8 + k];
  nodes[((long)b * NN + NI) * H + h] = acc > 0.f ? acc : 0.f;
}

// ---------------------------------------------------------------------------
// Kernel 3: kvec[h] = sum_h' wQKk[h'] * WK[h'][h]   (sk bias cancels in softmax)
// ---------------------------------------------------------------------------
__global__ void kvec_kernel(const float* __restrict__ WK,
                            const float* __restrict__ wQKk,
                            float* __restrict__ kvec)
{
  const int h = threadIdx.x;
  if (h < H) {
    float acc = 0.f;
    for (int hp = 0; hp < H; ++hp) acc += wQKk[hp] * WK[hp * H + h];
    kvec[h] = acc;
  }
}

// ---------------------------------------------------------------------------
// Kernel 4: sk[b*NN + j] = nodes[b][j] . kvec
// ---------------------------------------------------------------------------
__global__ void sk_kernel(const float* __restrict__ nodes,
                          const float* __restrict__ kvec,
                          float* __restrict__ sk)
{
  const int idx = blockIdx.x * blockDim.x + threadIdx.x;
  if (idx >= BB * NN) return;
  const float* row = nodes + (long)idx * H;
  float acc = 0.f;
  #pragma unroll 8
  for (int h = 0; h < H; ++h) acc += row[h] * kvec[h];
  sk[idx] = acc;
}

// ---------------------------------------------------------------------------
// Kernel 5: per-batch softmax stats: max and 1/sum(exp)
// ---------------------------------------------------------------------------
__global__ void stats_kernel(const float* __restrict__ sk,
                             float* __restrict__ mOut, float* __restrict__ invS)
{
  __shared__ float red[256];
  const int b = blockIdx.x, t = threadIdx.x;
  const float* s = sk + (long)b * NN;
  float mx = -1e30f;
  for (int j = t; j < NN; j += 256) mx = fmaxf(mx, s[j]);
  red[t] = mx; __syncthreads();
  for (int st = 128; st > 0; st >>= 1) {
    if (t < st) red[t] = fmaxf(red[t], red[t + st]);
    __syncthreads();
  }
  mx = red[0]; __syncthreads();
  float sm = 0.f;
  for (int j = t; j < NN; j += 256) sm += __expf(s[j] - mx);
  red[t] = sm; __syncthreads();
  for (int st = 128; st > 0; st >>= 1) {
    if (t < st) red[t] += red[t + st];
    __syncthreads();
  }
  if (t == 0) { mOut[b] = mx; invS[b] = 1.f / red[0]; }
}

// ---------------------------------------------------------------------------
// Kernel 6: pooled[b][h] = sum_j softmax_j(sk[b]) * nodes[b][j][h]
// 256 threads = 4 j-groups x 64 h lanes (coalesced over h)
// ---------------------------------------------------------------------------
__global__ void pool_kernel(const float* __restrict__ nodes,
                            const float* __restrict__ sk,
                            const float* __restrict__ mIn,
                            const float* __restrict__ invS,
                            float* __restrict__ pooled)
{
  __shared__ float part[4][H];
  const int b = blockIdx.x;
  const int h = threadIdx.x & 63;
  const int g = threadIdx.x >> 6;
  const float m = mIn[b];
  const float* nb = nodes + (long)b * NN * H;
  const float* sb = sk + (long)b * NN;
  float acc = 0.f;
  for (int j = g; j < NN; j += 4) acc += __expf(sb[j] - m) * nb[j * H + h];
  part[g][h] = acc; __syncthreads();
  if (g == 0)
    pooled[b * H + h] = (part[0][h] + part[1][h] + part[2][h] + part[3][h]) * invS[b];
}

// ---------------------------------------------------------------------------
// Kernel 7: 4 collapsed communicate rounds + heads. One block, 8 waves.
//   round r: Y = X @ WV^T + bV ; X = Y @ WA^T + bA ; BN over batch ; relu
//   (round 0 input X = pooled; rounds 1..3 attention is uniform over
//    identical nodes -> pooled == X itself)
// ---------------------------------------------------------------------------
__global__ void __launch_bounds__(256) head_kernel(
    const float* __restrict__ pooled,
    const float* __restrict__ WV, const float* __restrict__ bV,
    const float* __restrict__ WA, const float* __restrict__ bA,
    const float* __restrict__ gamma, const float* __restrict__ beta,
    const float* __restrict__ Wmu,  const float* __restrict__ bmu,
    const float* __restrict__ Wsig, const float* __restrict__ bsig,
    float* __restrict__ out)
{
  __shared__ float X[BB][H];
  __shared__ float Y[BB][H];
  __shared__ float scaleS[H], shiftS[H];
  const int t = threadIdx.x;
  const int lane = t & 31, wave = t >> 5;
  const int l15 = lane & 15, lhalf = lane >> 4;
  const int mtile = wave >> 2, ntile = wave & 3;   // 2 x 4 tiles of 32x64

  for (int i = t; i < BB * H; i += 256) X[i / H][i % H] = pooled[i];
  __syncthreads();

  for (int round = 0; round < 4; ++round) {
    { // Y = X @ WV^T + bV
      v8f c = {};
      #pragma unroll
      for (int s = 0; s < 16; ++s) {
        const int kb = 4 * s + 2 * lhalf;
        v2f a, bf;
        const int m = mtile * 16 + l15;
        a.x = X[m][kb]; a.y = X[m][kb + 1];
        const int nG = ntile * 16 + l15;
        bf.x = WV[nG * H + kb]; bf.y = WV[nG * H + kb + 1];
        c = wmma_f32_k4(a, bf, c);
      }
      #pragma unroll
      for (int r = 0; r < 8; ++r) {
        const int m  = mtile * 16 + r + 8 * lhalf;
        const int nG = ntile * 16 + l15;
        Y[m][nG] = c[r] + bV[nG];
      }
    }
    __syncthreads();
    { // X = Y @ WA^T + bA
      v8f c = {};
      #pragma unroll
      for (int s = 0; s < 16; ++s) {
        const int kb = 4 * s + 2 * lhalf;
        v2f a, bf;
        const int m = mtile * 16 + l15;
        a.x = Y[m][kb]; a.y = Y[m][kb + 1];
        const int nG = ntile * 16 + l15;
        bf.x = WA[nG * H + kb]; bf.y = WA[nG * H + kb + 1];
        c = wmma_f32_k4(a, bf, c);
      }
      #pragma unroll
      for (int r = 0; r < 8; ++r) {
        const int m  = mtile * 16 + r + 8 * lhalf;
        const int nG = ntile * 16 + l15;
        X[m][nG] = c[r] + bA[nG];
      }
    }
    __syncthreads();
    // BatchNorm over batch axis (biased var, matches jnp.var) + relu
    if (t < H) {
      float mu = 0.f, m2 = 0.f;
      for (int bI = 0; bI < BB; ++bI) { const float v = X[bI][t]; mu += v; m2 += v * v; }
      mu *= (1.f / BB); m2 *= (1.f / BB);
      const float var = m2 - mu * mu;
      const float sc = gamma[t] * rsqrtf(var + BN_EPS);
      scaleS[t] = sc;
      shiftS[t] = beta[t] - mu * sc;
    }
    __syncthreads();
    for (int i = t; i < BB * H; i += 256) {
      const int hI = i % H;
      const float v = X[i / H][hI] * scaleS[hI] + shiftS[hI];
      X[i / H][hI] = v > 0.f ? v : 0.f;
    }
    __syncthreads();
  }

  // heads: agg = max over identical nodes = X itself
  if (t < BB) {
    float mu = bmu[0], sg = bsig[0];
    for (int h = 0; h < H; ++h) { mu += X[t][h] * Wmu[h]; sg += X[t][h] * Wsig[h]; }
    out[t]      = mu;
    out[BB + t] = sg * sg + 0.01f;
  }
}

// ---------------------------------------------------------------------------
extern "C" void kernel_launch(void* const* d_in, const int* in_sizes, int n_in,
                              void* d_out, int out_size, void* d_ws, size_t ws_size,
                              hipStream_t stream) {
  (void)in_sizes; (void)n_in; (void)out_size; (void)ws_size;
  const float* xx    = (const float*)d_in[0];
  const float* yy    = (const float*)d_in[1];
  const float* oxx   = (const float*)d_in[2];
  const float* W1    = (const float*)d_in[3];
  const float* b1    = (const float*)d_in[4];
  const float* W2    = (const float*)d_in[5];
  const float* b2    = (const float*)d_in[6];
  const float* Wenc  = (const float*)d_in[7];
  const float* benc  = (const float*)d_in[8];
  // d_in[9]=WQ, [10]=bQ, [15]=wQKq, [17]=bQK unused: they cancel in softmax
  const float* WK    = (const float*)d_in[11];
  const float* WV    = (const float*)d_in[13];
  const float* bV    = (const float*)d_in[14];
  const float* wQKk  = (const float*)d_in[16];
  const float* WA    = (const float*)d_in[18];
  const float* bA    = (const float*)d_in[19];
  const float* gma   = (const float*)d_in[20];
  const float* beta  = (const float*)d_in[21];
  const float* Wmu   = (const float*)d_in[22];
  const float* bmu   = (const float*)d_in[23];
  const float* Wsig  = (const float*)d_in[24];
  const float* bsig  = (const float*)d_in[25];
  float* out = (float*)d_out;

  float* ws     = (float*)d_ws;
  float* nodes  = ws;                              // B*NN*H = 3,145,728 f
  float* sk     = nodes  + (size_t)BB * NN * H;    // B*NN   =    49,152 f
  float* kvec   = sk     + (size_t)BB * NN;        // H
  float* mBuf   = kvec   + H;                      // B
  float* invS   = mBuf   + BB;                     // B
  float* pooled = invS   + BB;                     // B*H

  encode_kernel<<<384, 256, 0, stream>>>(xx, yy, W1, b1, W2, b2, nodes);
  outenc_kernel<<<8, 256, 0, stream>>>(oxx, Wenc, benc, nodes);
  kvec_kernel<<<1, 64, 0, stream>>>(WK, wQKk, kvec);
  sk_kernel<<<(BB * NN + 255) / 256, 256, 0, stream>>>(nodes, kvec, sk);
  stats_kernel<<<BB, 256, 0, stream>>>(sk, mBuf, invS);
  pool_kernel<<<BB, 256, 0, stream>>>(nodes, sk, mBuf, invS, pooled);
  head_kernel<<<1, 256, 0, stream>>>(pooled, WV, bV, WA, bA, gma, beta,
                                     Wmu, bmu, Wsig, bsig, out);
}